// ContrastiveLearningModel_90151363543101
// MI455X (gfx1250) — compile-verified
//
#include <hip/hip_runtime.h>

#define NN 100000      // N_ENTITY
#define DD 128         // entity hidden
#define HIDN 256       // hidden_size
#define NREL 48
#define NBASE 8
#define NE 1000000     // edges
#define MAXU 1024      // B_SZ * E_LEN upper bound on unique targets

typedef __attribute__((ext_vector_type(2))) float v2f;
typedef __attribute__((ext_vector_type(8))) float v8f;

__device__ __forceinline__ v8f wmma4(v2f a, v2f b, v8f c) {
  // V_WMMA_F32_16X16X4_F32 : D = A(16x4) x B(4x16) + C, full f32
  return __builtin_amdgcn_wmma_f32_16x16x4_f32(false, a, false, b, (short)0, c,
                                               false, false);
}

// ---------------- init: map=-1, deg=0, acc=0, zrow=0, count=0 ----------------
__global__ void k_init(int* __restrict__ map, int* __restrict__ deg,
                       float* __restrict__ acc, float* __restrict__ zrow,
                       int* __restrict__ uniqc) {
  int i = blockIdx.x * blockDim.x + threadIdx.x;
  if (i < NN) map[i] = -1;
  if (i < MAXU * NREL) deg[i] = 0;
  if (i < MAXU * NBASE * DD) acc[i] = 0.f;
  if (i < DD) zrow[i] = 0.f;
  if (i == 0) *uniqc = 0;
}

// ---------------- unique target slots ----------------
__global__ void k_uniq(const int* __restrict__ entity_ids, int* __restrict__ map,
                       int* __restrict__ uniqn, int* __restrict__ uniqc) {
  int i = blockIdx.x * blockDim.x + threadIdx.x;
  if (i >= MAXU) return;
  int node = entity_ids[i];
  if (atomicCAS(&map[node], -1, -2) == -1) {
    int s = atomicAdd(uniqc, 1);
    uniqn[s] = node;
    atomicExch(&map[node], s);
  }
}

// ---------------- edge pass 1: per (slot, relation) in-degree ----------------
__global__ void k_deg(const int* __restrict__ eidx, const int* __restrict__ etype,
                      const int* __restrict__ map, int* __restrict__ deg) {
  int e = blockIdx.x * blockDim.x + threadIdx.x;
  if (e >= NE) return;
  if (e + 8192 < NE) {
    __builtin_prefetch(eidx + NE + e + 8192, 0, 0);
    __builtin_prefetch(etype + e + 8192, 0, 0);
  }
  int s = map[eidx[NE + e]];
  if (s >= 0) atomicAdd(&deg[s * NREL + etype[e]], 1);
}

// ---------------- edge pass 2: acc[slot][k][:] += comp[t,k]*norm * x[src] ----
__global__ void k_acc(const int* __restrict__ eidx, const int* __restrict__ etype,
                      const int* __restrict__ map, const int* __restrict__ deg,
                      const float* __restrict__ comp, const float* __restrict__ x,
                      float* __restrict__ acc) {
  int e = blockIdx.x * blockDim.x + threadIdx.x;
  int lane = threadIdx.x & 31;
  int slot = -1, sn = 0, t = 0;
  if (e < NE) {
    int s = map[eidx[NE + e]];
    if (s >= 0) { slot = s; sn = eidx[e]; t = etype[e]; }
  }
  unsigned long long m = __ballot(slot >= 0);
  while (m) {
    int sl = __ffsll(m) - 1;
    m &= m - 1;
    int s   = __shfl(slot, sl);
    int ssn = __shfl(sn, sl);
    int tt  = __shfl(t, sl);
    float nrm = 1.0f / fmaxf((float)deg[s * NREL + tt], 1.0f);
    const float4* xr = (const float4*)(x + (size_t)ssn * DD);
    float4 xv = xr[lane];                       // lane covers dims 4*lane..4*lane+3
    float* ab = acc + (size_t)s * (NBASE * DD) + lane * 4;
#pragma unroll
    for (int k = 0; k < NBASE; ++k) {
      float wk = comp[tt * NBASE + k] * nrm;
      float* p = ab + k * DD;
      atomicAdd(p + 0, wk * xv.x);
      atomicAdd(p + 1, wk * xv.y);
      atomicAdd(p + 2, wk * xv.z);
      atomicAdd(p + 3, wk * xv.w);
    }
  }
}

// ---------------- h = x_u@root + sum_k acc_k@bases_k + bias + x_u ------------
__global__ void __launch_bounds__(32)
k_h(const float* __restrict__ x, const float* __restrict__ root,
    const float* __restrict__ bases, const float* __restrict__ bias,
    const float* __restrict__ acc, const float* __restrict__ zrow,
    const int* __restrict__ uniqn, const int* __restrict__ uniqc,
    float* __restrict__ h) {
  const int U = *uniqc;
  const int row0 = blockIdx.x * 16;
  if (row0 >= U) return;
  const int col0 = blockIdx.y * 16;
  const int lane = threadIdx.x;
  const int half = lane >> 4;
  const int lm = lane & 15;
  const int aslot = row0 + lm;
  const bool av = aslot < U;
  // invalid rows read a zeroed scratch row -> no per-iteration predication
  const float2* xr2 =
      av ? (const float2*)(x + (size_t)uniqn[aslot] * DD) : (const float2*)zrow;
  v8f c = {};
#pragma unroll 16
  for (int k = 0; k < DD; k += 4) {               // x_u @ root
    float2 af = xr2[(k >> 1) + half];
    v2f a; a.x = af.x; a.y = af.y;
    v2f b;
    b.x = root[(k + 2 * half) * DD + col0 + lm];
    b.y = root[(k + 2 * half + 1) * DD + col0 + lm];
    c = wmma4(a, b, c);
  }
  for (int kb = 0; kb < NBASE; ++kb) {            // acc_k @ bases_k
    // acc rows for slots >= U are zero-initialized: no guard needed
    const float2* ar2 = (const float2*)(acc + ((size_t)aslot * NBASE + kb) * DD);
    const float* B = bases + (size_t)kb * DD * DD;
#pragma unroll 16
    for (int k = 0; k < DD; k += 4) {
      float2 af = ar2[(k >> 1) + half];
      v2f a; a.x = af.x; a.y = af.y;
      v2f b;
      b.x = B[(k + 2 * half) * DD + col0 + lm];
      b.y = B[(k + 2 * half + 1) * DD + col0 + lm];
      c = wmma4(a, b, c);
    }
  }
  float bb = bias[col0 + lm];
#pragma unroll
  for (int v = 0; v < 8; ++v) {
    int r = row0 + v + 8 * half;
    if (r < U) {
      int nd = uniqn[r];
      h[(size_t)r * DD + col0 + lm] =
          c[v] + bb + x[(size_t)nd * DD + col0 + lm];
    }
  }
}

// ---------------- h2 = relu(h@w1+b1)@w2 + b2 + h (fused, LDS intermediate) --
__global__ void __launch_bounds__(128)
k_mlp(const float* __restrict__ h, const float* __restrict__ w1,
      const float* __restrict__ b1, const float* __restrict__ w2,
      const float* __restrict__ b2, const int* __restrict__ uniqc,
      float* __restrict__ h2) {
  __shared__ float tbuf[16][64];
  const int U = *uniqc;
  const int row0 = blockIdx.x * 16;
  if (row0 >= U) return;
  const int wv = threadIdx.x >> 5;     // wave 0..3
  const int lane = threadIdx.x & 31;
  const int half = lane >> 4;
  const int lm = lane & 15;
  {
    const int col0 = wv * 16;          // 4 waves cover 64 cols of t
    const float2* ar2 = (const float2*)(h + (size_t)(row0 + lm) * DD);
    v8f c = {};
#pragma unroll 16
    for (int k = 0; k < DD; k += 4) {
      float2 af = ar2[(k >> 1) + half];
      v2f a; a.x = af.x; a.y = af.y;
      v2f b;
      b.x = w1[(k + 2 * half) * 64 + col0 + lm];
      b.y = w1[(k + 2 * half + 1) * 64 + col0 + lm];
      c = wmma4(a, b, c);
    }
    float bb = b1[col0 + lm];
#pragma unroll
    for (int v = 0; v < 8; ++v)
      tbuf[v + 8 * half][col0 + lm] = fmaxf(c[v] + bb, 0.f);
  }
  __syncthreads();
#pragma unroll
  for (int ct = 0; ct < 2; ++ct) {     // each wave: 2 output col-tiles
    const int col0 = (wv * 2 + ct) * 16;
    v8f c = {};
#pragma unroll 16
    for (int k = 0; k < 64; k += 4) {
      v2f a;
      a.x = tbuf[lm][k + 2 * half];
      a.y = tbuf[lm][k + 2 * half + 1];
      v2f b;
      b.x = w2[(k + 2 * half) * DD + col0 + lm];
      b.y = w2[(k + 2 * half + 1) * DD + col0 + lm];
      c = wmma4(a, b, c);
    }
    float bb = b2[col0 + lm];
#pragma unroll
    for (int v = 0; v < 8; ++v) {
      int r = row0 + v + 8 * half;
      if (r < U)
        h2[(size_t)r * DD + col0 + lm] =
            c[v] + bb + h[(size_t)r * DD + col0 + lm];
    }
  }
}

// ---------------- ent = h2 @ wp + bp ----------------
__global__ void __launch_bounds__(32)
k_proj(const float* __restrict__ h2, const float* __restrict__ wp,
       const float* __restrict__ bp, const int* __restrict__ uniqc,
       float* __restrict__ ent) {
  const int U = *uniqc;
  const int row0 = blockIdx.x * 16;
  if (row0 >= U) return;
  const int col0 = blockIdx.y * 16;
  const int lane = threadIdx.x;
  const int half = lane >> 4;
  const int lm = lane & 15;
  const float2* ar2 = (const float2*)(h2 + (size_t)(row0 + lm) * DD);
  v8f c = {};
#pragma unroll 16
  for (int k = 0; k < DD; k += 4) {
    float2 af = ar2[(k >> 1) + half];
    v2f a; a.x = af.x; a.y = af.y;
    v2f b;
    b.x = wp[(k + 2 * half) * HIDN + col0 + lm];
    b.y = wp[(k + 2 * half + 1) * HIDN + col0 + lm];
    c = wmma4(a, b, c);
  }
  float bb = bp[col0 + lm];
#pragma unroll
  for (int v = 0; v < 8; ++v) {
    int r = row0 + v + 8 * half;
    if (r < U) ent[(size_t)r * HIDN + col0 + lm] = c[v] + bb;
  }
}

// ---------------- out[b,l,:] = ent[slot(entity_ids[b,l]), :] ----------------
__global__ void k_gather(const float* __restrict__ ent, const int* __restrict__ map,
                         const int* __restrict__ entity_ids,
                         float* __restrict__ out) {
  int i = blockIdx.x * blockDim.x + threadIdx.x;
  if (i >= MAXU * HIDN) return;
  int c = i & (HIDN - 1);
  int e = i >> 8;
  int s = map[entity_ids[e]];
  out[i] = ent[(size_t)s * HIDN + c];
}

extern "C" void kernel_launch(void* const* d_in, const int* in_sizes, int n_in,
                              void* d_out, int out_size, void* d_ws, size_t ws_size,
                              hipStream_t stream) {
  (void)in_sizes; (void)n_in; (void)out_size; (void)ws_size;
  const float* node_embeds = (const float*)d_in[0];
  const float* bases       = (const float*)d_in[1];
  const float* comp        = (const float*)d_in[2];
  const float* root        = (const float*)d_in[3];
  const float* bias        = (const float*)d_in[4];
  const float* w1          = (const float*)d_in[5];
  const float* b1          = (const float*)d_in[6];
  const float* w2          = (const float*)d_in[7];
  const float* b2          = (const float*)d_in[8];
  const float* wp          = (const float*)d_in[9];
  const float* bp          = (const float*)d_in[10];
  const int*   edge_index  = (const int*)d_in[11];
  const int*   edge_type   = (const int*)d_in[12];
  const int*   entity_ids  = (const int*)d_in[13];
  float* out = (float*)d_out;

  char* ws = (char*)d_ws;
  size_t off = 0;
  auto take = [&](size_t bytes) -> void* {
    void* p = ws + off;
    off += (bytes + 255) & ~(size_t)255;
    return p;
  };
  int*   map   = (int*)take((size_t)NN * 4);
  int*   uniqc = (int*)take(4);
  int*   uniqn = (int*)take((size_t)MAXU * 4);
  int*   deg   = (int*)take((size_t)MAXU * NREL * 4);
  float* acc   = (float*)take((size_t)MAXU * NBASE * DD * 4);
  float* hbuf  = (float*)take((size_t)MAXU * DD * 4);
  float* h2buf = (float*)take((size_t)MAXU * DD * 4);
  float* ent   = (float*)take((size_t)MAXU * HIDN * 4);
  float* zrow  = (float*)take((size_t)DD * 4);

  k_init<<<(MAXU * NBASE * DD + 255) / 256, 256, 0, stream>>>(map, deg, acc,
                                                              zrow, uniqc);
  k_uniq<<<(MAXU + 255) / 256, 256, 0, stream>>>(entity_ids, map, uniqn, uniqc);
  k_deg<<<(NE + 255) / 256, 256, 0, stream>>>(edge_index, edge_type, map, deg);
  k_acc<<<(NE + 255) / 256, 256, 0, stream>>>(edge_index, edge_type, map, deg,
                                              comp, node_embeds, acc);
  k_h<<<dim3(MAXU / 16, DD / 16), 32, 0, stream>>>(node_embeds, root, bases, bias,
                                                   acc, zrow, uniqn, uniqc, hbuf);
  k_mlp<<<MAXU / 16, 128, 0, stream>>>(hbuf, w1, b1, w2, b2, uniqc, h2buf);
  k_proj<<<dim3(MAXU / 16, HIDN / 16), 32, 0, stream>>>(h2buf, wp, bp, uniqc, ent);
  k_gather<<<(MAXU * HIDN + 255) / 256, 256, 0, stream>>>(ent, map, entity_ids, out);
}